// AudioEvent_67224828117306
// MI455X (gfx1250) — compile-verified
//
#include <hip/hip_runtime.h>
#include <math.h>

// ---- module constants ----
#define SEQ_       64
#define N_SAMPLES_ 16384
#define N_EVENTS_  16
#define N_HARM_    32
#define NOISE_WS_  512
#define NBINS_     257
#define KPAD_      272          // 17 * 16 (bins padded to WMMA tile multiple)
#define BATCH_     8
#define TOTAL_OUT_ (BATCH_ * N_EVENTS_ * N_SAMPLES_)

#define MIN_F0_  (20.0f / 11025.0f)
#define F0DIFF_  (780.0f / 11025.0f)
#define PI_      3.14159265358979323846f
#define TWO_PI_  6.28318530717958647692f

typedef __attribute__((ext_vector_type(2))) float v2f;
typedef __attribute__((ext_vector_type(8))) float v8f;

// post-ERB fundamental, normalized [0,1] -> f0n in normalized freq
__device__ __forceinline__ float f0_post_erb(float fv, float f0b) {
    fv = fminf(fmaxf(fv, -0.5f), 0.5f);
    float erb = (0.108f * (f0b * 11025.0f) + 24.7f) * (1.0f / 11025.0f);
    float ff  = fminf(fmaxf(f0b + fv * erb, 0.0f), 1.0f);
    return MIN_F0_ + ff * F0DIFF_;
}

// linear-interp weights for sample n out of 16384 over a length-64 sequence
__device__ __forceinline__ void interp_w(int n, int& i0, int& i1, float& w) {
    float pos = ((float)n + 0.5f) * (1.0f / 256.0f) - 0.5f;
    pos = fminf(fmaxf(pos, 0.0f), 63.0f);
    i0 = (int)pos;              // pos >= 0 so truncation == floor
    i1 = min(i0 + 1, 63);
    w  = pos - (float)i0;
}

// ------------------------------------------------------------------
// Kernel 0: DFT basis tables.
//   fwd_cos/fwd_sin: [t=0..511][k=0..271]  cos/sin(2*pi*k*t/512)
//   inv_cos/inv_sin: [k=0..271][t=0..511]  same values, transposed layout
// ------------------------------------------------------------------
__global__ void gen_basis(float* __restrict__ fwd_cos, float* __restrict__ fwd_sin,
                          float* __restrict__ inv_cos, float* __restrict__ inv_sin) {
    int idx = blockIdx.x * blockDim.x + threadIdx.x;
    if (idx >= 512 * KPAD_) return;
    int t = idx / KPAD_;
    int k = idx - t * KPAD_;
    int m = (t * k) & 511;                       // exact modular reduction
    float ang = (float)m * (TWO_PI_ / 512.0f);
    float s, c;
    sincosf(ang, &s, &c);
    fwd_cos[t * KPAD_ + k] = c;
    fwd_sin[t * KPAD_ + k] = s;
    inv_cos[k * 512 + t]   = c;
    inv_sin[k * 512 + t]   = s;
}

// ------------------------------------------------------------------
// Kernel 1: per-frame windowed DFT -> Gaussian band filter -> iDFT.
// One workgroup = 16 frames (M dim of WMMA). 8 waves.
//   Forward: C[k]=sum_t xw[t]cos, S[k]=sum_t xw[t]sin   via v_wmma_f32_16x16x4_f32
//   Scale:   Z* = (c_k * gauss(k) / 512) * {C,S},  c_k = 1 for k in {0,256}, else 2
//   Inverse: y[t] = sum_k Zc[k]cos(2pi k t/512) + Zs[k]sin(2pi k t/512)
// A-frag layout (32b A 16x4): lanes0-15 M=0..15 K={t0,t0+1}; lanes16-31 K={t0+2,t0+3}
// B-frag layout (32b B 4x16): vgpr0 K=t0/t0+2, vgpr1 K=t0+1/t0+3, N = lane&15
// C/D layout: vgpr r -> M = r + 8*(lane>=16), N = lane&15
// ------------------------------------------------------------------
__device__ __forceinline__ void fwd_tile(const float* __restrict__ xw,
                                         const float* __restrict__ fwd_cos,
                                         const float* __restrict__ fwd_sin,
                                         int kt, int l16, int koff,
                                         v8f& accC, v8f& accS) {
    const int ncol = kt * 16 + l16;
    for (int t0 = 0; t0 < NOISE_WS_; t0 += 4) {
        v2f a;
        a.x = xw[l16 * 513 + t0 + koff];
        a.y = xw[l16 * 513 + t0 + koff + 1];
        int r0 = (t0 + koff) * KPAD_ + ncol;
        v2f bc, bs;
        bc.x = fwd_cos[r0];
        bc.y = fwd_cos[r0 + KPAD_];
        bs.x = fwd_sin[r0];
        bs.y = fwd_sin[r0 + KPAD_];
        accC = __builtin_amdgcn_wmma_f32_16x16x4_f32(false, a, false, bc, (short)0, accC, false, false);
        accS = __builtin_amdgcn_wmma_f32_16x16x4_f32(false, a, false, bs, (short)0, accS, false, false);
    }
}

__device__ __forceinline__ void write_tile(float* __restrict__ zc, float* __restrict__ zs,
                                           const float* __restrict__ meanL,
                                           const float* __restrict__ stdL,
                                           int kt, int l16, int half,
                                           v8f accC, v8f accS) {
    const int ncol = kt * 16 + l16;
    const float ck = (ncol == 0 || ncol == 256) ? 1.0f : 2.0f;
    const float fr = (float)ncol * (1.0f / 256.0f);
    const bool valid = (ncol < NBINS_);
#pragma unroll
    for (int r = 0; r < 8; r++) {
        int m = r + half * 8;
        float w = 0.0f;
        if (valid) {
            float dz = (fr - meanL[m]) / stdL[m];
            w = ck * __expf(-0.5f * dz * dz) * (1.0f / 512.0f);
        }
        zc[m * 273 + ncol] = accC[r] * w;
        zs[m * 273 + ncol] = accS[r] * w;
    }
}

__launch_bounds__(256)
__global__ void noise_dft(const float* __restrict__ noise_frames,
                          const float* __restrict__ f0,
                          const float* __restrict__ noise_std,
                          const float* __restrict__ f0_baselines,
                          const float* __restrict__ fwd_cos, const float* __restrict__ fwd_sin,
                          const float* __restrict__ inv_cos, const float* __restrict__ inv_sin,
                          float* __restrict__ y_frames) {
    // xw [16][513] (8208 floats) aliases zc/zs [16][273] x2 (8736 floats) after the
    // forward pass: accumulators are held in VGPRs across the barrier.
    __shared__ float buf[8736];
    __shared__ float meanL[16], stdL[16];

    const int tid = threadIdx.x;
    const int frameBase = blockIdx.x * 16;      // 16 frames per WG; 64 frames per (b,e)

    float* xw = buf;
    float* zc = buf;
    float* zs = buf + 4368;

    // load + Hann window 16x512 frames into LDS
    for (int i = tid; i < 16 * NOISE_WS_; i += 256) {
        int m = i >> 9, t = i & 511;
        float win = 0.5f - 0.5f * __cosf((float)t * (TWO_PI_ / 512.0f));
        xw[m * 513 + t] = noise_frames[(size_t)(frameBase + m) * NOISE_WS_ + t] * win;
    }
    if (tid < 16) {
        int g  = frameBase + tid;
        int s  = g & 63;
        int be = g >> 6;
        float f0b = f0_baselines[be];
        meanL[tid] = f0_post_erb(f0[(size_t)be * SEQ_ + s], f0b);
        float sd = fminf(fmaxf(noise_std[(size_t)be * SEQ_ + s], 1e-12f), 1.0f);
        stdL[tid] = sd * F0DIFF_;
    }
    // warm L2 for the inverse-basis streams used after the barrier
    for (int i = tid; i < KPAD_; i += 256) {
        __builtin_prefetch(inv_cos + (size_t)i * 512, 0, 1);
        __builtin_prefetch(inv_sin + (size_t)i * 512, 0, 1);
    }
    __syncthreads();

    const int lane = tid & 31;
    const int wave = tid >> 5;
    const int half = lane >> 4;
    const int l16  = lane & 15;
    const int koff = half * 2;

    // ---- forward DFT: 17 k-tiles over 8 waves (wave0 takes the 17th) ----
    v8f c0 = {}, s0 = {}, c1 = {}, s1 = {}, c2 = {}, s2 = {};
    fwd_tile(xw, fwd_cos, fwd_sin, wave,     l16, koff, c0, s0);
    fwd_tile(xw, fwd_cos, fwd_sin, wave + 8, l16, koff, c1, s1);
    if (wave == 0) fwd_tile(xw, fwd_cos, fwd_sin, 16, l16, koff, c2, s2);
    __syncthreads();                       // done reading xw; LDS re-used as spectra

    write_tile(zc, zs, meanL, stdL, wave,     l16, half, c0, s0);
    write_tile(zc, zs, meanL, stdL, wave + 8, l16, half, c1, s1);
    if (wave == 0) write_tile(zc, zs, meanL, stdL, 16, l16, half, c2, s2);
    __syncthreads();

    // ---- inverse DFT: 32 t-tiles over 8 waves, K = 272 (zeros beyond bin 256) ----
    for (int tt = wave; tt < 32; tt += 8) {
        const int tcol = tt * 16 + l16;
        v8f acc = {};
        for (int k0 = 0; k0 < KPAD_; k0 += 4) {
            v2f ac, asn;
            ac.x  = zc[l16 * 273 + k0 + koff];
            ac.y  = zc[l16 * 273 + k0 + koff + 1];
            asn.x = zs[l16 * 273 + k0 + koff];
            asn.y = zs[l16 * 273 + k0 + koff + 1];
            int r0 = (k0 + koff) * 512 + tcol;
            v2f bc, bs;
            bc.x = inv_cos[r0];
            bc.y = inv_cos[r0 + 512];
            bs.x = inv_sin[r0];
            bs.y = inv_sin[r0 + 512];
            acc = __builtin_amdgcn_wmma_f32_16x16x4_f32(false, ac,  false, bc, (short)0, acc, false, false);
            acc = __builtin_amdgcn_wmma_f32_16x16x4_f32(false, asn, false, bs, (short)0, acc, false, false);
        }
#pragma unroll
        for (int r = 0; r < 8; r++) {
            int m = r + half * 8;
            y_frames[(size_t)(frameBase + m) * NOISE_WS_ + tcol] = acc[r];
        }
    }
}

// ------------------------------------------------------------------
// Kernel 2: overlap-add (hop 256) + noise_mix; fully overwrites d_out.
//   sample n of (b,e): frame fa = n>>8 first half [n&255], frame fa-1 second half.
// ------------------------------------------------------------------
__global__ void combine_noise(const float* __restrict__ y_frames,
                              const float* __restrict__ overall_env,
                              float* __restrict__ out) {
    int idx = blockIdx.x * blockDim.x + threadIdx.x;
    if (idx >= TOTAL_OUT_) return;
    int n  = idx & (N_SAMPLES_ - 1);
    int be = idx >> 14;
    int fa = n >> 8;
    int ta = n & 255;
    size_t gf = (size_t)be * SEQ_ + fa;
    float noise = y_frames[gf * NOISE_WS_ + ta];
    if (fa > 0) noise += y_frames[(gf - 1) * NOISE_WS_ + 256 + ta];
    int i0, i1; float w;
    interp_w(n, i0, i1, w);
    float e0 = 1.0f - fminf(fmaxf(overall_env[(size_t)be * SEQ_ + i0], 0.0f), 1.0f);
    float e1 = 1.0f - fminf(fmaxf(overall_env[(size_t)be * SEQ_ + i1], 0.0f), 1.0f);
    out[idx] = noise * (e0 + (e1 - e0) * w);
}

// ------------------------------------------------------------------
// Kernel 3: oscillator bank. One WG per (b, e, track), track 0 = fundamental,
// tracks 1..32 = harmonics (factor h+1). 256 threads x 64 samples each.
// Phase = pi * cumsum(interp(freq)); block-level inclusive scan over thread
// partials; per-thread phase offsets range-reduced mod 2*pi so the running
// phase stays within ~78 revolutions -> safe domain for hardware V_SIN_F32.
// sin * env * osc_mix accumulated into d_out with f32 atomics.
// ------------------------------------------------------------------
__launch_bounds__(256)
__global__ void osc_accum(const float* __restrict__ f0,
                          const float* __restrict__ overall_env,
                          const float* __restrict__ osc_env,
                          const float* __restrict__ harm_env,
                          const float* __restrict__ f0_baselines,
                          float* __restrict__ out) {
    __shared__ float fseq[64], eseq[64], mseq[64];
    __shared__ float part[256];

    const int tid = threadIdx.x;
    const int ch  = blockIdx.x;
    const int h   = ch % 33;          // 0 = fundamental
    const int be  = ch / 33;

    if (tid < 64) {
        int s = tid;
        float f0n = f0_post_erb(f0[(size_t)be * SEQ_ + s], f0_baselines[be]);
        float fac = (h == 0) ? 1.0f : (float)(h + 1);
        fseq[s] = f0n * fac * PI_;                      // phase increment
        float oe = fminf(fmaxf(osc_env[(size_t)be * SEQ_ + s], 0.0f), 1.0f);
        float env = oe;
        if (h > 0) {
            float he = fminf(fmaxf(harm_env[((size_t)be * N_HARM_ + (h - 1)) * SEQ_ + s], 0.0f), 1.0f);
            env = oe * he;
        }
        eseq[s] = env;
        mseq[s] = fminf(fmaxf(overall_env[(size_t)be * SEQ_ + s], 0.0f), 1.0f);
    }
    __syncthreads();

    // pass 1: per-thread partial sum of the 64 phase increments it owns
    const int base = tid * 64;
    float sum = 0.0f;
    for (int j = 0; j < 64; j++) {
        int i0, i1; float w;
        interp_w(base + j, i0, i1, w);
        sum += fseq[i0] + (fseq[i1] - fseq[i0]) * w;
    }
    part[tid] = sum;
    __syncthreads();

    // inclusive Hillis-Steele scan over 256 partials
    for (int st = 1; st < 256; st <<= 1) {
        float v = (tid >= st) ? part[tid - st] : 0.0f;
        __syncthreads();
        part[tid] += v;
        __syncthreads();
    }
    float off = part[tid] - sum;                        // exclusive prefix
    off -= TWO_PI_ * floorf(off * (1.0f / TWO_PI_));    // range-reduce

    // pass 2: running phase, hardware sine, env, mix, accumulate
    float phase = off;
    float* outc = out + (size_t)be * N_SAMPLES_;
    for (int j = 0; j < 64; j++) {
        int n = base + j;
        int i0, i1; float w;
        interp_w(n, i0, i1, w);
        phase += fseq[i0] + (fseq[i1] - fseq[i0]) * w;
        float env = eseq[i0] + (eseq[i1] - eseq[i0]) * w;
        float mix = mseq[i0] + (mseq[i1] - mseq[i0]) * w;
        atomicAdd(&outc[n], __sinf(phase) * env * mix); // V_SIN_F32: |phase| < 256 rev
    }
}

// ------------------------------------------------------------------
extern "C" void kernel_launch(void* const* d_in, const int* in_sizes, int n_in,
                              void* d_out, int out_size, void* d_ws, size_t ws_size,
                              hipStream_t stream) {
    (void)in_sizes; (void)n_in; (void)out_size; (void)ws_size;
    const float* f0           = (const float*)d_in[0];
    const float* overall_env  = (const float*)d_in[1];
    const float* osc_env      = (const float*)d_in[2];
    /* d_in[3] = noise_env : unused by the reference */
    const float* harm_env     = (const float*)d_in[4];
    const float* noise_std    = (const float*)d_in[5];
    const float* f0_baselines = (const float*)d_in[6];
    const float* noise_frames = (const float*)d_in[7];
    float* out = (float*)d_out;

    // workspace: 4 basis tables (512*272 each) + per-frame irfft output (8192*512)
    const size_t basisElems = (size_t)512 * KPAD_;
    float* fwd_cos  = (float*)d_ws;
    float* fwd_sin  = fwd_cos + basisElems;
    float* inv_cos  = fwd_sin + basisElems;
    float* inv_sin  = inv_cos + basisElems;
    float* y_frames = inv_sin + basisElems;   // 8192 * 512 floats

    int basisThreads = (int)basisElems;
    gen_basis<<<(basisThreads + 255) / 256, 256, 0, stream>>>(fwd_cos, fwd_sin, inv_cos, inv_sin);

    noise_dft<<<BATCH_ * N_EVENTS_ * SEQ_ / 16, 256, 0, stream>>>(
        noise_frames, f0, noise_std, f0_baselines,
        fwd_cos, fwd_sin, inv_cos, inv_sin, y_frames);

    combine_noise<<<(TOTAL_OUT_ + 255) / 256, 256, 0, stream>>>(y_frames, overall_env, out);

    osc_accum<<<BATCH_ * N_EVENTS_ * 33, 256, 0, stream>>>(
        f0, overall_env, osc_env, harm_env, f0_baselines, out);
}